// LocalTransportPath_9835475108490
// MI455X (gfx1250) — compile-verified
//
#include <hip/hip_runtime.h>
#include <hip/hip_bf16.h>
#include <math.h>

// Problem constants (from the reference)
#define BQ 16
#define HQ 336
#define CQ 512
#define KQ 8
#define MTOT (BQ * HQ)      // 5376 rows
#define KDIM CQ             // 512 reduction dim
#define NTOT (CQ * 2 * KQ)  // 8192 GEMM columns
#define CLIPV 6.0f
#define DOMAINV 12.0f

// Tiling
#define BM 32               // rows of raw per block (2 wmma M-subtiles)
#define BCG 4               // spline groups (of 16 cols) per block
#define BN (BCG * 16)       // 64 N columns per block
#define BK 32               // K chunk staged in LDS

#define A_STRIDE 36         // 32 + pad, keeps float4 alignment, kills bank conflicts
#define B_STRIDE 65         // 64 + pad
#define R_STRIDE 17         // 16 + pad for the raw-tile spill

typedef __attribute__((ext_vector_type(2))) float v2f;
typedef __attribute__((ext_vector_type(8))) float v8f;

__global__ __launch_bounds__(256, 2)
void fused_spline_gemm_kernel(const float* __restrict__ fst,    // (M, 512) future_state_time
                              const float* __restrict__ ybase,  // (M, 512)
                              const float* __restrict__ Wm,     // (8192, 512), row o over c
                              const float* __restrict__ bias,   // (8192)
                              float* __restrict__ out)          // (M, 512)
{
    __shared__ float Alds[BM * A_STRIDE];          // 4.6 KB
    __shared__ float Blds[BK * B_STRIDE];          // 8.3 KB  (stored as [k][n])
    __shared__ float Rlds[8 * 16 * R_STRIDE];      // 8.7 KB  (raw tiles for epilogue)

    const int tid  = threadIdx.x;
    const int lane = tid & 31;
    const int w    = tid >> 5;        // wave id 0..7
    const int mt   = w & 1;           // M sub-tile (0..1)
    const int nt   = w >> 1;          // N sub-tile / c-group in block (0..3)
    const int half = lane >> 4;       // half-wave: selects K pair per ISA A/B layout
    const int l16  = lane & 15;

    const int m0  = blockIdx.y * BM;  // global row base
    const int cg0 = blockIdx.x * BCG; // global c-group base
    const int n0  = cg0 * 16;         // global N base

    // Bias depends only on the column; each lane owns one column of the D tile.
    const float bcol = bias[n0 + nt * 16 + l16];

    v8f acc = {};

    for (int k0 = 0; k0 < KDIM; k0 += BK) {
        // ---- Stage A chunk (BM x BK) : 1024 floats, one float4 per thread ----
        {
            const int r  = tid >> 3;            // 0..31 row
            const int kk = (tid & 7) << 2;      // 0..28 k
            const float4 va = *(const float4*)(&fst[(size_t)(m0 + r) * KDIM + k0 + kk]);
            *(float4*)(&Alds[r * A_STRIDE + kk]) = va;
        }
        // ---- Stage B chunk: 64 W-rows x BK, transposed into Blds[k][n] ----
        #pragma unroll
        for (int it = 0; it < 2; ++it) {
            const int idx = tid + it * 256;     // 0..511 float4 slots
            const int n   = idx >> 3;           // 0..63
            const int kk  = (idx & 7) << 2;     // 0..28
            const float4 vb = *(const float4*)(&Wm[(size_t)(n0 + n) * KDIM + k0 + kk]);
            Blds[(kk + 0) * B_STRIDE + n] = vb.x;
            Blds[(kk + 1) * B_STRIDE + n] = vb.y;
            Blds[(kk + 2) * B_STRIDE + n] = vb.z;
            Blds[(kk + 3) * B_STRIDE + n] = vb.w;
        }
        __syncthreads();

        // ---- 8 x V_WMMA_F32_16X16X4_F32 per chunk ----
        // ISA 32-bit A 16x4 layout: lanes 0-15 hold K={0,1}, lanes 16-31 hold K={2,3}.
        #pragma unroll
        for (int kk = 0; kk < BK; kk += 4) {
            const float* ap = &Alds[(mt * 16 + l16) * A_STRIDE + kk + 2 * half];
            v2f afrag; afrag.x = ap[0]; afrag.y = ap[1];

            const float* bp = &Blds[(kk + 2 * half) * B_STRIDE + nt * 16 + l16];
            v2f bfrag; bfrag.x = bp[0]; bfrag.y = bp[B_STRIDE];

            acc = __builtin_amdgcn_wmma_f32_16x16x4_f32(
                /*neg_a=*/false, afrag, /*neg_b=*/false, bfrag,
                /*c_mod=*/(short)0, acc, /*reuse_a=*/false, /*reuse_b=*/false);
        }
        __syncthreads();
    }

    // ---- Add bias and spill this wave's 16x16 raw tile to LDS ----
    // C/D layout: VGPR g, lane L -> row (g + 8*(L>>4)), col (L&15).
    #pragma unroll
    for (int g = 0; g < 8; ++g) {
        const int row = g + 8 * half;
        Rlds[(w * 16 + row) * R_STRIDE + l16] = acc[g] + bcol;
    }
    __syncthreads();

    // ---- Epilogue: 128 lanes, one (row m, group c) each ----
    if (tid < 128) {
        const int t = tid >> 4;        // tile 0..7  (mt = t&1, nt = t>>1)
        const int r = tid & 15;        // row within tile
        const float* rp = &Rlds[(t * 16 + r) * R_STRIDE];

        // softmax over raw_w[0..7] -> widths (scaled to DOMAIN)
        float ew[8];
        float mx = rp[0];
        #pragma unroll
        for (int j = 1; j < 8; ++j) mx = fmaxf(mx, rp[j]);
        float sum = 0.0f;
        #pragma unroll
        for (int j = 0; j < 8; ++j) { ew[j] = __expf(rp[j] - mx); sum += ew[j]; }
        const float scale = DOMAINV / sum;

        float sl[8];
        #pragma unroll
        for (int j = 0; j < 8; ++j) sl[j] = __expf(rp[8 + j]);

        const int m = m0 + (t & 1) * 16 + r;
        const int c = cg0 + (t >> 1);
        const float x = ybase[(size_t)m * CQ + c];

        // Running cumsum of knots + running bin select:
        // idx = max j in [0,7] with x >= x_bounds[j] (else 0)
        float xcur = -CLIPV, ycur = -CLIPV;
        float xl = xcur, yl = ycur, sk = sl[0];
        #pragma unroll
        for (int j = 0; j < 8; ++j) {
            if (j == 0 || x >= xcur) { xl = xcur; yl = ycur; sk = sl[j]; }
            const float wdt = ew[j] * scale;
            xcur += wdt;
            ycur += sl[j] * wdt;
        }
        out[(size_t)m * CQ + c] = yl + sk * (x - xl);
    }
}

extern "C" void kernel_launch(void* const* d_in, const int* in_sizes, int n_in,
                              void* d_out, int out_size, void* d_ws, size_t ws_size,
                              hipStream_t stream) {
    // Input order from setup_inputs():
    // 0 y_norm, 1 y_base, 2 future_state_hat, 3 future_state_time,
    // 4 mu_base_fut, 5 std_base_fut, 6 base_time_mean, 7 base_time_std, 8 W, 9 b
    const float* ybase = (const float*)d_in[1];
    const float* fst   = (const float*)d_in[3];
    const float* Wm    = (const float*)d_in[8];
    const float* bias  = (const float*)d_in[9];
    float* out = (float*)d_out;

    dim3 grid(CQ / BCG, MTOT / BM);   // (128, 168)
    dim3 block(256);
    fused_spline_gemm_kernel<<<grid, block, 0, stream>>>(fst, ybase, Wm, bias, out);
}